// GCN_60258391163406
// MI455X (gfx1250) — compile-verified
//
#include <hip/hip_runtime.h>

typedef __attribute__((ext_vector_type(2))) float v2f;
typedef __attribute__((ext_vector_type(8))) float v8f;

// ---------------------------------------------------------------------------
// Generic float4 zero-fill
// ---------------------------------------------------------------------------
__global__ __launch_bounds__(256) void zero_f4(float* __restrict__ p, int n4) {
    int i = blockIdx.x * blockDim.x + threadIdx.x;
    if (i < n4) {
        *(float4*)(p + (size_t)i * 4) = make_float4(0.f, 0.f, 0.f, 0.f);
    }
}

// ---------------------------------------------------------------------------
// Degree count: deg[dst[e]] += 1  (float atomics; deg buffer pre-zeroed)
// ---------------------------------------------------------------------------
__global__ __launch_bounds__(256) void deg_count(const int* __restrict__ dst,
                                                 float* __restrict__ deg, int nedges) {
    int e = blockIdx.x * blockDim.x + threadIdx.x;
    if (e < nedges) atomicAdd(&deg[dst[e]], 1.0f);
}

// ---------------------------------------------------------------------------
// In place: dinv[i] = rsqrt(deg[i] + 1)   (+1 = self loop)
// ---------------------------------------------------------------------------
__global__ __launch_bounds__(256) void rsqrt_k(float* __restrict__ d, int n) {
    int i = blockIdx.x * blockDim.x + threadIdx.x;
    if (i < n) d[i] = rsqrtf(d[i] + 1.0f);
}

// ---------------------------------------------------------------------------
// H = X @ W using native f32 WMMA (V_WMMA_F32_16X16X4_F32).
// Block = 256 threads = 8 waves; block covers 128 rows x all 128 cols.
// W (128x128) staged in LDS with stride 132 (bank-conflict padding).
// Wave w: rows r0 = blockIdx*128 + w*16.  A fragments (32 k-steps x v2f)
// held in registers; 8 column tiles of 16 swept per wave.
//
// f32 WMMA operand layout (ISA 7.12.2, 32-bit A 16x4):
//   lane = 16*half + lid :  A vgpr0 = A[M=lid][K = 2*half+0], vgpr1 = K=2*half+1
//   B (4x16):              B vgpr0 = B[K=2*half+0][N=lid],   vgpr1 = K=2*half+1
//   C/D vgpr v:            D[M = v + 8*half][N = lid]
// ---------------------------------------------------------------------------
__global__ __launch_bounds__(256) void gemm_wmma(const float* __restrict__ X,
                                                 const float* __restrict__ W,
                                                 float* __restrict__ H, int nrows) {
    __shared__ float sW[128 * 132];

    // cooperative load of W into LDS (padded stride 132)
    for (int it = 0; it < 16; ++it) {
        int idx4 = threadIdx.x + it * 256;       // 4096 float4 total
        int row  = idx4 >> 5;
        int col  = (idx4 & 31) << 2;
        *(float4*)&sW[row * 132 + col] = *(const float4*)&W[row * 128 + col];
    }
    __syncthreads();

    const int wave = threadIdx.x >> 5;
    const int lane = threadIdx.x & 31;
    const int lid  = lane & 15;
    const int half = lane >> 4;

    const int r0 = blockIdx.x * 128 + wave * 16;
    if (r0 >= nrows) return;                     // wave-uniform: EXEC stays full

    // A fragments: row r0+lid, columns (4*ks + 2*half) .. +1
    int arow = r0 + lid;
    if (arow >= nrows) arow = nrows - 1;         // clamp (N is a multiple of 16)
    const float* xrow = X + (size_t)arow * 128 + 2 * half;

    v2f areg[32];
#pragma unroll
    for (int ks = 0; ks < 32; ++ks) {
        areg[ks] = *(const v2f*)(xrow + ks * 4);
    }

    for (int ct = 0; ct < 8; ++ct) {
        const int cb = ct * 16 + lid;
        v8f acc = {0.f, 0.f, 0.f, 0.f, 0.f, 0.f, 0.f, 0.f};
#pragma unroll
        for (int ks = 0; ks < 32; ++ks) {
            const int kk = ks * 4 + 2 * half;
            v2f b;
            b.x = sW[kk * 132 + cb];
            b.y = sW[kk * 132 + 132 + cb];
            acc = __builtin_amdgcn_wmma_f32_16x16x4_f32(
                false, areg[ks], false, b, (short)0, acc, false, false);
        }
        float* hp = H + (size_t)(r0 + 8 * half) * 128 + cb;
#pragma unroll
        for (int v = 0; v < 8; ++v) {
            int row = r0 + 8 * half + v;
            if (row < nrows) hp[(size_t)v * 128] = acc[v];
        }
    }
}

// ---------------------------------------------------------------------------
// Edge messages: agg[dst] += h[src] * dinv[src]*dinv[dst].
// One wave per edge: 32 lanes x float4 = 128 features, coalesced gather,
// scatter via global_atomic_add_f32 (agg buffer is L2-resident: 51 MB << 192 MB).
// ---------------------------------------------------------------------------
__global__ __launch_bounds__(256) void edge_msg(const float* __restrict__ H,
                                                const float* __restrict__ dinv,
                                                const int* __restrict__ src,
                                                const int* __restrict__ dst,
                                                float* __restrict__ agg, int nedges) {
    int e = (blockIdx.x * blockDim.x + threadIdx.x) >> 5;
    if (e >= nedges) return;
    const int lane = threadIdx.x & 31;
    const int s = src[e];
    const int d = dst[e];
    const float norm = dinv[s] * dinv[d];
    const float4 hv = *(const float4*)(H + (size_t)s * 128 + lane * 4);
    float* ap = agg + (size_t)d * 128 + lane * 4;
    atomicAdd(ap + 0, hv.x * norm);
    atomicAdd(ap + 1, hv.y * norm);
    atomicAdd(ap + 2, hv.z * norm);
    atomicAdd(ap + 3, hv.w * norm);
}

// ---------------------------------------------------------------------------
// agg = relu(agg + h * dinv^2 + bias)   (in place on agg, float4 elementwise)
// ---------------------------------------------------------------------------
__global__ __launch_bounds__(256) void combine_k(float* __restrict__ agg,
                                                 const float* __restrict__ H,
                                                 const float* __restrict__ dinv,
                                                 const float* __restrict__ bias, int n4) {
    int i4 = blockIdx.x * blockDim.x + threadIdx.x;
    if (i4 >= n4) return;
    const int node = i4 >> 5;        // 32 float4 per node (128 feats)
    const int f4   = (i4 & 31) * 4;
    const float di = dinv[node];
    const float sc = di * di;
    float4 a = *(float4*)(agg + (size_t)i4 * 4);
    const float4 h = *(const float4*)(H + (size_t)i4 * 4);
    const float4 b = *(const float4*)(bias + f4);
    a.x = fmaxf(fmaf(h.x, sc, a.x) + b.x, 0.f);
    a.y = fmaxf(fmaf(h.y, sc, a.y) + b.y, 0.f);
    a.z = fmaxf(fmaf(h.z, sc, a.z) + b.z, 0.f);
    a.w = fmaxf(fmaf(h.w, sc, a.w) + b.w, 0.f);
    *(float4*)(agg + (size_t)i4 * 4) = a;
}

// ---------------------------------------------------------------------------
// Pool: sums[batch[n]] += h[n]; counts[batch[n]] += 1. One wave per node.
// ---------------------------------------------------------------------------
__global__ __launch_bounds__(256) void pool_k(const float* __restrict__ H,
                                              const int* __restrict__ batch,
                                              float* __restrict__ sums,
                                              float* __restrict__ counts, int nnodes) {
    int node = (blockIdx.x * blockDim.x + threadIdx.x) >> 5;
    if (node >= nnodes) return;
    const int lane = threadIdx.x & 31;
    const int g = batch[node];
    const float4 hv = *(const float4*)(H + (size_t)node * 128 + lane * 4);
    float* sp = sums + (size_t)g * 128 + lane * 4;
    atomicAdd(sp + 0, hv.x);
    atomicAdd(sp + 1, hv.y);
    atomicAdd(sp + 2, hv.z);
    atomicAdd(sp + 3, hv.w);
    if (lane == 0) atomicAdd(&counts[g], 1.0f);
}

// ---------------------------------------------------------------------------
// out[g] = (sums[g]/max(counts[g],1)) . w_out + b_out.  One wave per graph.
// ---------------------------------------------------------------------------
__global__ __launch_bounds__(256) void final_k(const float* __restrict__ sums,
                                               const float* __restrict__ counts,
                                               const float* __restrict__ wout,
                                               const float* __restrict__ bout,
                                               float* __restrict__ out, int ngraphs) {
    int g = (blockIdx.x * blockDim.x + threadIdx.x) >> 5;
    if (g >= ngraphs) return;
    const int lane = threadIdx.x & 31;
    const float inv = 1.0f / fmaxf(counts[g], 1.0f);
    const float4 s = *(const float4*)(sums + (size_t)g * 128 + lane * 4);
    const float4 w = *(const float4*)(wout + lane * 4);
    float acc = (s.x * w.x + s.y * w.y) + (s.z * w.z + s.w * w.w);
    acc *= inv;
#pragma unroll
    for (int off = 16; off > 0; off >>= 1) acc += __shfl_xor(acc, off, 32);
    if (lane == 0) out[g] = acc + bout[0];
}

// ---------------------------------------------------------------------------
extern "C" void kernel_launch(void* const* d_in, const int* in_sizes, int n_in,
                              void* d_out, int out_size, void* d_ws, size_t ws_size,
                              hipStream_t stream) {
    (void)n_in; (void)ws_size;

    const float* x     = (const float*)d_in[0];
    const int*   eidx  = (const int*)d_in[1];
    const int*   batch = (const int*)d_in[2];
    const float* w1    = (const float*)d_in[3];
    const float* b1    = (const float*)d_in[4];
    const float* w2    = (const float*)d_in[5];
    const float* b2    = (const float*)d_in[6];
    const float* wout  = (const float*)d_in[7];
    const float* bout  = (const float*)d_in[8];
    float*       out   = (float*)d_out;

    const int N = in_sizes[0] / 128;   // 100000 nodes
    const int E = in_sizes[1] / 2;     // 1.6M edges
    const int G = out_size;            // 256 graphs

    const int* src = eidx;
    const int* dst = eidx + E;

    // workspace layout
    float* bufA   = (float*)d_ws;                                  // N*128 (GEMM out)
    float* bufB   = bufA + (size_t)N * 128;                        // N*128 (agg / h)
    float* dinv   = bufB + (size_t)N * 128;                        // N
    float* sums   = dinv + (((size_t)N + 255) & ~(size_t)255);     // G*128
    float* counts = sums + (size_t)G * 128;                        // G

    const int TB = 256;
    const int nf   = N * 32;                        // float4 count of a feature buf
    const int gElt = (N + 127) / 128;               // GEMM blocks (128 rows each)
    const int gEdge = (E * 32 + TB - 1) / TB;       // wave per edge
    const int gNode = (N * 32 + TB - 1) / TB;       // wave per node

    // ---- degree / normalization -------------------------------------------
    zero_f4<<<((N + 3) / 4 + TB - 1) / TB, TB, 0, stream>>>(dinv, (N + 3) / 4);
    deg_count<<<(E + TB - 1) / TB, TB, 0, stream>>>(dst, dinv, E);
    rsqrt_k<<<(N + TB - 1) / TB, TB, 0, stream>>>(dinv, N);

    // ---- layer 1 -----------------------------------------------------------
    gemm_wmma<<<gElt, TB, 0, stream>>>(x, w1, bufA, N);
    zero_f4<<<(nf + TB - 1) / TB, TB, 0, stream>>>(bufB, nf);
    edge_msg<<<gEdge, TB, 0, stream>>>(bufA, dinv, src, dst, bufB, E);
    combine_k<<<(nf + TB - 1) / TB, TB, 0, stream>>>(bufB, bufA, dinv, b1, nf);

    // ---- layer 2 -----------------------------------------------------------
    gemm_wmma<<<gElt, TB, 0, stream>>>(bufB, w2, bufA, N);
    zero_f4<<<(nf + TB - 1) / TB, TB, 0, stream>>>(bufB, nf);
    edge_msg<<<gEdge, TB, 0, stream>>>(bufA, dinv, src, dst, bufB, E);
    combine_k<<<(nf + TB - 1) / TB, TB, 0, stream>>>(bufB, bufA, dinv, b2, nf);

    // ---- pool + readout ----------------------------------------------------
    const int pz4 = (G * 128 + G + 3) / 4;          // sums + counts contiguous
    zero_f4<<<(pz4 + TB - 1) / TB, TB, 0, stream>>>(sums, pz4);
    pool_k<<<gNode, TB, 0, stream>>>(bufB, batch, sums, counts, N);
    final_k<<<(G * 32 + TB - 1) / TB, TB, 0, stream>>>(sums, counts, wout, bout, out, G);
}